// ProtoBoundaryLoss_54150947668077
// MI455X (gfx1250) — compile-verified
//
#include <hip/hip_runtime.h>
#include <hip/hip_bf16.h>
#include <math.h>

typedef __attribute__((ext_vector_type(2))) float v2f;
typedef __attribute__((ext_vector_type(8))) float v8f;
typedef __attribute__((ext_vector_type(4))) unsigned int v4u;
typedef __attribute__((ext_vector_type(4))) int v4i;
typedef __attribute__((ext_vector_type(8))) int v8i;

#define EMB 512
#define NCL 4096
#define BSTRIDE 516              // padded LDS row stride in floats (TDM pad: +2 dw per 256 dw)
#define TILE_N 32                // columns staged per block iteration (2 wmma tiles)
#define NT (NCL / TILE_N)        // 128 iterations
#define ROWS_PER_BLOCK 128       // 8 waves x 16 rows
#define NEG_BIG -3.402823466e38f
#define TILE_FLOATS (TILE_N * BSTRIDE)
#define TILE_BYTES (TILE_FLOATS * 4)

#if __has_builtin(__builtin_amdgcn_tensor_load_to_lds)
#define HAVE_TDM 1
#define NBUF 2
#else
#define HAVE_TDM 0
#define NBUF 1
#endif

#if HAVE_TDM
// ---------------------------------------------------------------------------
// TDM: DMA W[nb..nb+31, 0..511] (f32, row-major, stride 512) into LDS at
// lds_byte_off, padding +2 DWORDs every 256 DWORDs -> row stride 516 floats.
// D# per CDNA5 ISA (group0 128b, group1 256b, 2D tile).
// ---------------------------------------------------------------------------
__device__ __forceinline__ void tdm_load_wtile(const float* Wg, int nb,
                                               unsigned lds_byte_off) {
    unsigned long long ga =
        (unsigned long long)(const void*)(Wg + (size_t)nb * EMB);
    v4u g0;
    g0.x = 1u;                                          // count=1 (valid user D#)
    g0.y = lds_byte_off;                                // lds_addr [63:32]
    g0.z = (unsigned)(ga & 0xffffffffu);                // global_addr [95:64]
    g0.w = (unsigned)((ga >> 32) & 0x01ffffffu)         // global_addr [120:96]
           | (2u << 30);                                // type=2 ("image")
    v8i g1;
    g1[0] = (int)((2u << 16)        // data_size = 4B
                  | (1u << 20)      // pad_enable
                  | (7u << 22)      // pad_interval: 256 DWORDs
                  | (1u << 25));    // pad_amount: 2 DWORDs
    g1[1] = (int)((unsigned)(EMB & 0xffff) << 16);      // tensor_dim0[15:0]
    g1[2] = (int)((unsigned)(NCL & 0xffff) << 16);      // tensor_dim1[15:0]
    g1[3] = (int)((unsigned)EMB << 16);                 // tile_dim0 = 512
    g1[4] = (int)TILE_N;                                // tile_dim1 = 32
    g1[5] = (int)EMB;                                   // tensor_dim0_stride = 512
    g1[6] = 0;
    g1[7] = 0;
    v4i z4 = (v4i)0;
#if __has_include(<hip/amd_detail/amd_gfx1250_TDM.h>)
    __builtin_amdgcn_tensor_load_to_lds(g0, g1, z4, z4, (v8i)0, 0);
#else
    __builtin_amdgcn_tensor_load_to_lds(g0, g1, z4, z4, 0);
#endif
}
#endif

// Manual fallback staging: same kinked 516-stride layout, float2 stores (8B ok).
__device__ __forceinline__ void stage_manual(float* buf, const float* __restrict__ Wg,
                                             int nb, int tid) {
#pragma unroll
    for (int i = 0; i < 32; ++i) {
        int f2 = i * 256 + tid;            // 8192 float2 chunks (32 rows x 256)
        int n  = f2 >> 8;
        int k2 = f2 & 255;
        int k  = k2 * 2;
        float2 w2v = ((const float2*)(Wg + (size_t)(nb + n) * EMB))[k2];
        int ko = k + ((k >= 256) ? 2 : 0);
        *(float2*)&buf[n * BSTRIDE + ko] = w2v;
    }
}

// K-loop: one 16-row A strip vs two 16-col B tiles -> two accumulators.
__device__ __forceinline__ void gemm_tile(const float* __restrict__ zrow,
                                          const float* buf, int lm, int half,
                                          v8f& acc0, v8f& acc1) {
    acc0 = (v8f)0.f;
    acc1 = (v8f)0.f;
    const float* b0 = buf + lm * BSTRIDE + 2 * half;
    const float* b1 = buf + (lm + 16) * BSTRIDE + 2 * half;
#pragma unroll 8
    for (int kb = 0; kb < 256; kb += 4) {
        v2f a  = *(const v2f*)(zrow + kb);
        v2f w0 = *(const v2f*)(b0 + kb);
        v2f w1 = *(const v2f*)(b1 + kb);
        acc0 = __builtin_amdgcn_wmma_f32_16x16x4_f32(false, a, false, w0, (short)0, acc0, false, false);
        acc1 = __builtin_amdgcn_wmma_f32_16x16x4_f32(false, a, false, w1, (short)0, acc1, false, false);
    }
#pragma unroll 8
    for (int kb = 256; kb < 512; kb += 4) {           // +2 float kink after 256 dw
        v2f a  = *(const v2f*)(zrow + kb);
        v2f w0 = *(const v2f*)(b0 + kb + 2);
        v2f w1 = *(const v2f*)(b1 + kb + 2);
        acc0 = __builtin_amdgcn_wmma_f32_16x16x4_f32(false, a, false, w0, (short)0, acc0, false, false);
        acc1 = __builtin_amdgcn_wmma_f32_16x16x4_f32(false, a, false, w1, (short)0, acc1, false, false);
    }
}

// ---------------------------------------------------------------------------
// Pass 1: labels[m] = argmax_n ( z_t[m,:] @ W[n,:] + b[n] )
// ---------------------------------------------------------------------------
__global__ __launch_bounds__(256) void argmax_kernel(const float* __restrict__ Z,
                                                     const float* __restrict__ W,
                                                     const float* __restrict__ bias,
                                                     int* __restrict__ labels) {
    __shared__ __align__(16) float Bs[NBUF][TILE_FLOATS];

    const int tid  = threadIdx.x;
    const int wave = tid >> 5;
    const int lane = tid & 31;
    const int lm   = lane & 15;
    const int half = lane >> 4;
    const int m0   = blockIdx.x * ROWS_PER_BLOCK + wave * 16;

    const float* zrow = Z + (size_t)(m0 + lm) * EMB + 2 * half;

    float best[8];
    int   bidx[8];
#pragma unroll
    for (int r = 0; r < 8; ++r) { best[r] = NEG_BIG; bidx[r] = 0; }

#if HAVE_TDM
    unsigned lds_base = (unsigned)(unsigned long long)(void*)&Bs[0][0];
    if (wave == 0) {
        tdm_load_wtile(W, 0, lds_base);
        __builtin_amdgcn_s_wait_tensorcnt(0);
    }
    __syncthreads();
#endif

    for (int it = 0; it < NT; ++it) {
        const int nb = it * TILE_N;
#if HAVE_TDM
        if ((it + 1 < NT) && wave == 0)
            tdm_load_wtile(W, nb + TILE_N, lds_base + (unsigned)(((it + 1) & 1) * TILE_BYTES));
        const float* buf = &Bs[it & 1][0];
#else
        stage_manual(&Bs[0][0], W, nb, tid);
        __syncthreads();
        const float* buf = &Bs[0][0];
#endif
        v8f a0, a1;
        gemm_tile(zrow, buf, lm, half, a0, a1);

        const float bv0 = bias[nb + lm];
        const float bv1 = bias[nb + 16 + lm];
        const int   c0  = nb + lm;
        const int   c1  = nb + 16 + lm;
#pragma unroll
        for (int r = 0; r < 8; ++r) {
            float v0 = a0[r] + bv0;
            if (v0 > best[r]) { best[r] = v0; bidx[r] = c0; }
            float v1 = a1[r] + bv1;
            if (v1 > best[r]) { best[r] = v1; bidx[r] = c1; }
        }
#if HAVE_TDM
        if (wave == 0) __builtin_amdgcn_s_wait_tensorcnt(0);
#endif
        __syncthreads();
    }

#pragma unroll
    for (int mask = 1; mask < 16; mask <<= 1) {
#pragma unroll
        for (int r = 0; r < 8; ++r) {
            float ov = __shfl_xor(best[r], mask, 32);
            int   oi = __shfl_xor(bidx[r], mask, 32);
            if (ov > best[r] || (ov == best[r] && oi < bidx[r])) { best[r] = ov; bidx[r] = oi; }
        }
    }
    if (lm == 0) {
#pragma unroll
        for (int r = 0; r < 8; ++r)
            labels[m0 + half * 8 + r] = bidx[r];
    }
}

// ---------------------------------------------------------------------------
// Pass 2/3: per-block sum of CE(row) = logsumexp(row) - logit[row,label]
// ---------------------------------------------------------------------------
__global__ __launch_bounds__(256) void ce_kernel(const float* __restrict__ Z,
                                                 const float* __restrict__ W,
                                                 const float* __restrict__ bias,
                                                 const int* __restrict__ labels,
                                                 int use_rowlabel,
                                                 float* __restrict__ partials) {
    __shared__ __align__(16) float Bs[NBUF][TILE_FLOATS];
    __shared__ float red[16];

    const int tid  = threadIdx.x;
    const int wave = tid >> 5;
    const int lane = tid & 31;
    const int lm   = lane & 15;
    const int half = lane >> 4;
    const int m0   = blockIdx.x * ROWS_PER_BLOCK + wave * 16;

    const float* zrow = Z + (size_t)(m0 + lm) * EMB + 2 * half;

    int lab[8];
#pragma unroll
    for (int r = 0; r < 8; ++r) {
        int row = m0 + half * 8 + r;
        lab[r] = use_rowlabel ? (row >> 3) : labels[row];
    }

    float mx[8], ss[8], picked[8];
#pragma unroll
    for (int r = 0; r < 8; ++r) { mx[r] = NEG_BIG; ss[r] = 0.f; picked[r] = 0.f; }

#if HAVE_TDM
    unsigned lds_base = (unsigned)(unsigned long long)(void*)&Bs[0][0];
    if (wave == 0) {
        tdm_load_wtile(W, 0, lds_base);
        __builtin_amdgcn_s_wait_tensorcnt(0);
    }
    __syncthreads();
#endif

    for (int it = 0; it < NT; ++it) {
        const int nb = it * TILE_N;
#if HAVE_TDM
        if ((it + 1 < NT) && wave == 0)
            tdm_load_wtile(W, nb + TILE_N, lds_base + (unsigned)(((it + 1) & 1) * TILE_BYTES));
        const float* buf = &Bs[it & 1][0];
#else
        stage_manual(&Bs[0][0], W, nb, tid);
        __syncthreads();
        const float* buf = &Bs[0][0];
#endif
        v8f a0, a1;
        gemm_tile(zrow, buf, lm, half, a0, a1);

        const float bv0 = bias[nb + lm];
        const float bv1 = bias[nb + 16 + lm];
        const int   c0  = nb + lm;
        const int   c1  = nb + 16 + lm;
#pragma unroll
        for (int r = 0; r < 8; ++r) {
            float x0 = a0[r] + bv0;
            float x1 = a1[r] + bv1;
            if (c0 == lab[r]) picked[r] = x0;
            if (c1 == lab[r]) picked[r] = x1;
            if (x0 > mx[r]) { ss[r] = ss[r] * __expf(mx[r] - x0) + 1.0f; mx[r] = x0; }
            else            { ss[r] += __expf(x0 - mx[r]); }
            if (x1 > mx[r]) { ss[r] = ss[r] * __expf(mx[r] - x1) + 1.0f; mx[r] = x1; }
            else            { ss[r] += __expf(x1 - mx[r]); }
        }
#if HAVE_TDM
        if (wave == 0) __builtin_amdgcn_s_wait_tensorcnt(0);
#endif
        __syncthreads();
    }

#pragma unroll
    for (int mask = 1; mask < 16; mask <<= 1) {
#pragma unroll
        for (int r = 0; r < 8; ++r) {
            float om = __shfl_xor(mx[r], mask, 32);
            float os = __shfl_xor(ss[r], mask, 32);
            float op = __shfl_xor(picked[r], mask, 32);
            picked[r] += op;
            float nm = fmaxf(mx[r], om);
            ss[r] = ss[r] * __expf(mx[r] - nm) + os * __expf(om - nm);
            mx[r] = nm;
        }
    }

    if (lm == 0) {
        float s = 0.f;
#pragma unroll
        for (int r = 0; r < 8; ++r)
            s += (mx[r] + __logf(ss[r])) - picked[r];
        red[wave * 2 + half] = s;
    }
    __syncthreads();
    if (tid == 0) {
        float s = 0.f;
#pragma unroll
        for (int i = 0; i < 16; ++i) s += red[i];
        partials[blockIdx.x] = s;
    }
}

// ---------------------------------------------------------------------------
// Final: out = sum(enc partials)/8192 + sum(proto partials)/32768
// ---------------------------------------------------------------------------
__global__ __launch_bounds__(128) void reduce_kernel(const float* __restrict__ partials,
                                                     float* __restrict__ out) {
    __shared__ float se[128], sp[128];
    int t = threadIdx.x;
    float e = (t < 64) ? partials[t] : 0.f;
    float p = partials[64 + t] + partials[192 + t];
    se[t] = e; sp[t] = p;
    __syncthreads();
    for (int s = 64; s > 0; s >>= 1) {
        if (t < s) { se[t] += se[t + s]; sp[t] += sp[t + s]; }
        __syncthreads();
    }
    if (t == 0) out[0] = se[0] / 8192.0f + sp[0] / 32768.0f;
}

extern "C" void kernel_launch(void* const* d_in, const int* in_sizes, int n_in,
                              void* d_out, int out_size, void* d_ws, size_t ws_size,
                              hipStream_t stream) {
    const float* z_t   = (const float*)d_in[0];   // [8192, 512]
    const float* z_s   = (const float*)d_in[1];   // [8192, 512]
    const float* W     = (const float*)d_in[2];   // [4096, 512]
    const float* b     = (const float*)d_in[3];   // [4096]
    const float* queue = (const float*)d_in[4];   // [32768, 512]
    float* out = (float*)d_out;

    int*   labels = (int*)d_ws;                          // 8192 ints
    float* parts  = (float*)((char*)d_ws + 8192 * 4);    // 320 floats

    argmax_kernel<<<8192 / ROWS_PER_BLOCK, 256, 0, stream>>>(z_t, W, b, labels);
    ce_kernel<<<8192 / ROWS_PER_BLOCK, 256, 0, stream>>>(z_s, W, b, labels, 0, parts);
    ce_kernel<<<32768 / ROWS_PER_BLOCK, 256, 0, stream>>>(queue, W, b, nullptr, 1, parts + 64);
    reduce_kernel<<<1, 128, 0, stream>>>(parts, out);
}